// MaskSampler_58626303590903
// MI455X (gfx1250) — compile-verified
//
#include <hip/hip_runtime.h>
#include <stdint.h>

// ---------------- problem constants (from the reference) ----------------
#define NLAYERS 12
#define NH      12
#define VT      158           // V_TOTAL vertices
#define PT      32936         // P_TOTAL edges per row
#define LMB     32778         // _LM_BASE
#define NPOSW   1030          // ceil(PT/32)
#define SMEM_BYTES (PT*4 + (NPOSW+2)*4 + 32)   // masks row + pos bitset(+guards) + F

__device__ constexpr int LBASE[NLAYERS] = {
    0, 86, 653, 1701, 3230, 5240, 7731, 10703, 14156, 18090, 22505, 27401};

// ---------------- CDNA5 TDM (Tensor Data Mover) -------------------------
typedef unsigned int u32x4 __attribute__((ext_vector_type(4)));
typedef int          i32x4 __attribute__((ext_vector_type(4)));
typedef int          i32x8 __attribute__((ext_vector_type(8)));

#if __has_builtin(__builtin_amdgcn_tensor_load_to_lds) && \
    __has_builtin(__builtin_amdgcn_tensor_store_from_lds) && \
    __has_builtin(__builtin_amdgcn_s_wait_tensorcnt)
#define HAVE_TDM 1
#endif

// Build a 2D D# describing one contiguous PT*4-byte row (tile 32936 x 1, f32).
__device__ __forceinline__ void tdm_build(unsigned lds_addr, const void* gptr,
                                          u32x4& g0, i32x8& g1) {
  unsigned long long ga = (unsigned long long)(uintptr_t)gptr;
  g0[0] = 1u;                                         // count=1, user mode
  g0[1] = lds_addr;                                   // LDS byte address
  g0[2] = (unsigned)ga;                               // global addr [31:0]
  g0[3] = (unsigned)((ga >> 32) & 0x01FFFFFFull)      // global addr [56:32]
          | (2u << 30);                               // type = 2 ("image")
  const unsigned td0 = (unsigned)PT;                  // tensor_dim0 (elements)
  g1[0] = (int)(2u << 16);                            // data_size=2 -> 4 bytes
  g1[1] = (int)((td0 & 0xFFFFu) << 16);               // tensor_dim0[15:0] @63:48
  g1[2] = (int)((td0 >> 16) | (1u << 16));            // td0[31:16], tensor_dim1=1
  g1[3] = (int)((td0 & 0xFFFFu) << 16);               // tile_dim0 @127:112
  g1[4] = 1;                                          // tile_dim1=1, tile_dim2=0
  g1[5] = (int)td0;                                   // tensor_dim0_stride[31:0]
  g1[6] = 0;
  g1[7] = 0;
}

__device__ __forceinline__ void tdm_load(unsigned lds_addr, const void* g) {
#ifdef HAVE_TDM
  u32x4 g0; i32x8 g1; tdm_build(lds_addr, g, g0, g1);
  i32x4 z4 = (i32x4)0;
#if defined(__clang_major__) && __clang_major__ >= 23
  i32x8 z8 = (i32x8)0;
  __builtin_amdgcn_tensor_load_to_lds(g0, g1, z4, z4, z8, 0);
#else
  __builtin_amdgcn_tensor_load_to_lds(g0, g1, z4, z4, 0);
#endif
#endif
}

__device__ __forceinline__ void tdm_store(unsigned lds_addr, void* g) {
#ifdef HAVE_TDM
  u32x4 g0; i32x8 g1; tdm_build(lds_addr, g, g0, g1);
  i32x4 z4 = (i32x4)0;
#if defined(__clang_major__) && __clang_major__ >= 23
  i32x8 z8 = (i32x8)0;
  __builtin_amdgcn_tensor_store_from_lds(g0, g1, z4, z4, z8, 0);
#else
  __builtin_amdgcn_tensor_store_from_lds(g0, g1, z4, z4, 0);
#endif
#endif
}

// ---------------- helpers ------------------------------------------------
__device__ __forceinline__ unsigned ballot32(bool p) {
#if __has_builtin(__builtin_amdgcn_ballot_w32)
  return __builtin_amdgcn_ballot_w32(p);
#else
  return (unsigned)__ballot(p);
#endif
}

// extract 32 bits of the LDS bitset starting at arbitrary bit offset
__device__ __forceinline__ unsigned ext32(const unsigned* pos, int bit) {
  unsigned w0 = pos[bit >> 5];
  unsigned w1 = pos[(bit >> 5) + 1];
  int sh = bit & 31;
  return sh ? ((w0 >> sh) | (w1 << (32 - sh))) : w0;
}
__device__ __forceinline__ unsigned cmask(int rem) {
  return rem >= 32 ? 0xFFFFFFFFu : ((1u << rem) - 1u);
}

// ---------------- kernel: one workgroup per batch row --------------------
__global__ void __launch_bounds__(256)
mask_prune_kernel(const float* __restrict__ sp,
                  const float* __restrict__ unif,
                  float* __restrict__ out) {
  extern __shared__ unsigned char smem[];
  float*    mrow = (float*)smem;                       // PT floats (unif -> masks)
  unsigned* pos  = (unsigned*)(smem + (size_t)PT * 4u);// NPOSW words + 2 guards
  unsigned* Fsh  = pos + (NPOSW + 2);                  // 5 words: F = R & O

  const int tid = threadIdx.x;
  const size_t roff = (size_t)blockIdx.x * (size_t)PT;

  // ---- Stage 1: DMA the unif row into LDS (TDM) ----
#ifdef HAVE_TDM
  if (tid < 32) {
    tdm_load((unsigned)(uintptr_t)mrow, (const void*)(unif + roff));
    __builtin_amdgcn_s_wait_tensorcnt(0);
  }
#else
  for (int p = tid; p < PT; p += 256) mrow[p] = unif[roff + p];
#endif
  __syncthreads();

  // ---- Stage 2: elementwise mask in place + wave32 ballot bit-pack ----
  for (int p = tid; p < PT; p += 256) {
    float s  = sp[p];
    float pr = 1.0f / (1.0f + __expf(-s));
    float w  = pr * (1.0f - pr);
    float x  = (pr - mrow[p]) / w + 0.5f;
    float m  = fminf(fmaxf(x, 0.0f), 1.0f);
    mrow[p]  = m;
    unsigned bal = ballot32(m > 0.0f);
    if ((tid & 31) == 0) pos[p >> 5] = bal;
  }
  if (tid == 0) { pos[NPOSW] = 0u; pos[NPOSW + 1] = 0u; }
  __syncthreads();

  // ---- Stage 3: reachability passes, done by wave 0 (lane-parallel rows) ----
  if (tid < 32) {
    unsigned R[6] = {3u, 0u, 0u, 0u, 0u, 0u};   // vertices 0,1 reachable

    // forward
#pragma unroll
    for (int L = 0; L < NLAYERS; ++L) {
      const int Vq = 2 + 13 * L, Vm = Vq + 12, a0 = Vq;
      const int b = LBASE[L], mb = b + 36 * Vq;
      unsigned bal0 = 0u, bal1 = 0u;
#pragma unroll
      for (int pass = 0; pass < 2; ++pass) {
        int r = tid + 32 * pass;                 // (s,h) row = s*12+h
        unsigned acc = 0u;
        if (r < 36) {
          int s0 = b + r * Vq;
#pragma unroll
          for (int c = 0; c * 32 < Vq; ++c)
            acc |= ext32(pos, s0 + 32 * c) & R[c] & cmask(Vq - 32 * c);
        }
        unsigned blt = ballot32(acc != 0u);      // uniform across wave
        if (pass == 0) bal0 = blt; else bal1 = blt;
      }
      auto T = [&](int r) -> unsigned {
        return (r < 32) ? ((bal0 >> r) & 1u) : ((bal1 >> (r - 32)) & 1u);
      };
      unsigned ar = 0u;
#pragma unroll
      for (int h = 0; h < 12; ++h)
        ar |= (T(h) & T(12 + h) & T(24 + h)) << h;
      { // insert head bits BEFORE the mm scan (reference ordering)
        const int w = a0 >> 5, sh = a0 & 31;
        R[w] |= ar << sh;
        if (sh + 12 > 32) R[w + 1] |= ar >> (32 - sh);
      }
      unsigned accm = 0u;
#pragma unroll
      for (int c = 0; c * 32 < Vm; ++c)
        accm |= ext32(pos, mb + 32 * c) & R[c] & cmask(Vm - 32 * c);
      if (accm) R[(a0 + 12) >> 5] |= 1u << ((a0 + 12) & 31);
    }

    // backward (first prune folded in by AND-ing R everywhere)
    unsigned O[6] = {0u, 0u, 0u, 0u, 0u, 0u};
#pragma unroll
    for (int w = 0; w < 5; ++w)
      O[w] = ext32(pos, LMB + 32 * w) & R[w] & cmask(VT - 32 * w);

#pragma unroll
    for (int L = NLAYERS - 1; L >= 0; --L) {
      const int Vq = 2 + 13 * L, Vm = Vq + 12, a0 = Vq;
      const int b = LBASE[L], mb = b + 36 * Vq;
      const int vb = a0 + 12;                     // mm output vertex
      if (((R[vb >> 5] & O[vb >> 5]) >> (vb & 31)) & 1u) {
#pragma unroll
        for (int c = 0; c * 32 < Vm; ++c)
          O[c] |= ext32(pos, mb + 32 * c) & R[c] & cmask(Vm - 32 * c);
      }
      unsigned g;                                 // head gates (R&O)[a0+h]
      {
        const int w = a0 >> 5, sh = a0 & 31;
        unsigned f0 = R[w] & O[w], f1 = R[w + 1] & O[w + 1];
        unsigned fx = sh ? ((f0 >> sh) | (f1 << (32 - sh))) : f0;
        g = fx & 0x0FFFu;
      }
      unsigned acc[5] = {0u, 0u, 0u, 0u, 0u};
#pragma unroll
      for (int pass = 0; pass < 2; ++pass) {
        int r = tid + 32 * pass;
        if (r < 36) {
          int h = r % 12;
          if ((g >> h) & 1u) {
            int s0 = b + r * Vq;
#pragma unroll
            for (int c = 0; c * 32 < Vq; ++c)
              acc[c] |= ext32(pos, s0 + 32 * c) & cmask(Vq - 32 * c);
          }
        }
      }
#pragma unroll
      for (int c = 0; c * 32 < Vq; ++c) {        // OR-reduce across the wave
        unsigned v = acc[c];
        v |= __shfl_xor(v, 16, 32);
        v |= __shfl_xor(v, 8, 32);
        v |= __shfl_xor(v, 4, 32);
        v |= __shfl_xor(v, 2, 32);
        v |= __shfl_xor(v, 1, 32);
        O[c] |= v & R[c];
      }
    }
    if (tid == 0) {
#pragma unroll
      for (int w = 0; w < 5; ++w) Fsh[w] = R[w] & O[w];
    }
  }
  __syncthreads();

  // ---- Stage 4: zero non-kept edges in LDS: keep = F[out] & F[in] ----
#pragma unroll
  for (int L = 0; L < NLAYERS; ++L) {
    const int Vq = 2 + 13 * L, Vm = Vq + 12, a0 = Vq;
    const int b = LBASE[L], slab = 36 * Vq, mb = b + slab;
    for (int idx = tid; idx < slab; idx += 256) {
      int r = idx / Vq;                          // Vq literal after unroll
      int j = idx - r * Vq;
      int ov = a0 + (r % 12);
      unsigned keep = (Fsh[ov >> 5] >> (ov & 31)) & (Fsh[j >> 5] >> (j & 31)) & 1u;
      if (!keep) mrow[b + idx] = 0.0f;
    }
    const int vm = a0 + 12;
    const unsigned ob = (Fsh[vm >> 5] >> (vm & 31)) & 1u;
    for (int j = tid; j < Vm; j += 256)
      if (!(ob & (Fsh[j >> 5] >> (j & 31)) & 1u)) mrow[mb + j] = 0.0f;
  }
  for (int j = tid; j < VT; j += 256)
    if (!((Fsh[j >> 5] >> (j & 31)) & 1u)) mrow[LMB + j] = 0.0f;
  __syncthreads();

  // ---- Stage 5: DMA the finished row back to HBM (TDM) ----
#ifdef HAVE_TDM
  if (tid < 32) {
    tdm_store((unsigned)(uintptr_t)mrow, (void*)(out + roff));
    __builtin_amdgcn_s_wait_tensorcnt(0);
  }
#else
  for (int p = tid; p < PT; p += 256) out[roff + p] = mrow[p];
#endif
}

// ---------------- host entry ---------------------------------------------
extern "C" void kernel_launch(void* const* d_in, const int* in_sizes, int n_in,
                              void* d_out, int out_size, void* d_ws, size_t ws_size,
                              hipStream_t stream) {
  (void)n_in; (void)out_size; (void)d_ws; (void)ws_size;
  const float* sp   = (const float*)d_in[0];   // sample_params (PT)
  const float* unif = (const float*)d_in[1];   // unif (bz, PT)
  float* out = (float*)d_out;
  const int bz = in_sizes[1] / PT;

  // 135.9 KB dynamic LDS (2 workgroups per 320 KB WGP); deterministic per call.
  (void)hipFuncSetAttribute((const void*)mask_prune_kernel,
                            hipFuncAttributeMaxDynamicSharedMemorySize,
                            (int)SMEM_BYTES);
  mask_prune_kernel<<<dim3(bz), dim3(256), SMEM_BYTES, stream>>>(sp, unif, out);
}